// LearnableTopoGNN_85134841741353
// MI455X (gfx1250) — compile-verified
//
#include <hip/hip_runtime.h>

// Problem constants (match reference)
#define B_ 8
#define K_ 20000
#define C_ 64
#define E_ 320000
#define NEGV -1e9f

typedef float v2f __attribute__((ext_vector_type(2)));
typedef float v8f __attribute__((ext_vector_type(8)));

// ---- order-preserving float <-> uint32 key ----
__device__ __forceinline__ unsigned int fkey(float f) {
    unsigned int b = __float_as_uint(f);
    return (b & 0x80000000u) ? ~b : (b | 0x80000000u);
}
__device__ __forceinline__ float unfkey(unsigned int k) {
    unsigned int b = (k & 0x80000000u) ? (k ^ 0x80000000u) : ~k;
    return __uint_as_float(b);
}

// ---- zero the top-k slot array (sentinel 0 = empty; any real key > 0) ----
__global__ void zero_slots_kernel(unsigned long long* __restrict__ slots, long long n) {
    long long i = (long long)blockIdx.x * blockDim.x + threadIdx.x;
    if (i < n) slots[i] = 0ull;
}

// ---- y[b,k,c] = sum_j x[b,k,j] * W[c,j]  via V_WMMA_F32_16X16X4_F32 ----
// Block = 256 threads = 8 waves. Wave w: rows rb = 32*rblk + (w/4)*16, cols cb = (w%4)*16.
// A frag (16x4 f32): lane m=lane&15, hi=lane>>4; v0/v1 = x[rb+m][k0+2*hi +0/+1]
// B frag (4x16 f32): n=lane&15; v0/v1 = B[2*hi +0/+1][n] = W[cb+n][k0+2*hi +0/+1]
// D (16x16 f32): vgpr i -> row rb + hi*8 + i, col cb + (lane&15)
__global__ void gemm_phi_kernel(const float* __restrict__ x,
                                const float* __restrict__ W,
                                float* __restrict__ y) {
    int wave = threadIdx.x >> 5;
    int lane = threadIdx.x & 31;
    int b    = blockIdx.x / 625;        // 20000/32 = 625 row-blocks per graph
    int rblk = blockIdx.x % 625;
    int rb = rblk * 32 + (wave >> 2) * 16;
    int cb = (wave & 3) * 16;
    const float* xg = x + (long long)b * K_ * C_;
    float*       yg = y + (long long)b * K_ * C_;
    int m  = lane & 15;
    int hi = lane >> 4;

    v8f acc = {};
#pragma unroll
    for (int k0 = 0; k0 < C_; k0 += 4) {
        const float* ap = xg + (long long)(rb + m) * C_ + k0 + 2 * hi;
        v2f a; a.x = ap[0]; a.y = ap[1];
        const float* wp = W + (long long)(cb + m) * C_ + k0 + 2 * hi;
        v2f bm; bm.x = wp[0]; bm.y = wp[1];
        acc = __builtin_amdgcn_wmma_f32_16x16x4_f32(
            /*neg_a=*/false, a, /*neg_b=*/false, bm,
            /*c_mod=*/(short)0, acc, /*reuse_a=*/false, /*reuse_b=*/false);
    }
#pragma unroll
    for (int i = 0; i < 8; ++i) {
        int row = rb + hi * 8 + i;
        yg[(long long)row * C_ + cb + m] = acc[i];
    }
}

// ---- per-(b,dst) top-8 via sorted atomicMax cascade on packed (key, ~eid) ----
__global__ void edge_topk_kernel(const int* __restrict__ edges,
                                 const float* __restrict__ logits,
                                 const int* __restrict__ ntime,
                                 unsigned long long* __restrict__ slots) {
    long long i = (long long)blockIdx.x * blockDim.x + threadIdx.x;
    if (i >= (long long)B_ * E_) return;
    int b = (int)(i / E_);
    int e = (int)(i % E_);
    const int* eb = edges + (long long)b * 2 * E_;
    int src = eb[e];
    int dst = eb[E_ + e];
    const int* tb = ntime + (long long)b * K_;
    float lm = (tb[dst] >= tb[src]) ? logits[(long long)b * E_ + e] : NEGV;
    unsigned long long v =
        ((unsigned long long)fkey(lm) << 32) | (unsigned int)(~(unsigned int)e);
    unsigned long long* s = slots + ((long long)b * K_ + dst) * 8;
    // Slots only grow (atomicMax) -> if v <= current 8th value it can never be top-8.
    if (v <= s[7]) return;
#pragma unroll
    for (int j = 0; j < 8; ++j) {
        unsigned long long old = atomicMax(&s[j], v);
        v = (old < v) ? old : v;     // carry displaced value down
        if (v == 0ull) break;        // displaced the empty sentinel: done
    }
}

// ---- fused per-dst softmax + weighted gather + residual ----
// 64 threads per node (one per channel), 4 nodes per 256-thread block.
__global__ void aggregate_kernel(const float* __restrict__ x,
                                 const int* __restrict__ edges,
                                 const unsigned long long* __restrict__ slots,
                                 const float* __restrict__ y,
                                 float* __restrict__ out) {
    __shared__ float sw[4][8];
    __shared__ int   ssrc[4][8];
    int g = threadIdx.x >> 6;          // node slot in block: 0..3
    int c = threadIdx.x & 63;          // channel
    long long n = (long long)blockIdx.x * 4 + g;
    bool active = n < (long long)B_ * K_;
    int b = 0, k = 0;
    if (active) { b = (int)(n / K_); k = (int)(n % K_); (void)k; }

    if (active && c == 0) {
        const unsigned long long* s = slots + n * 8;
        float lv[8]; int eid[8];
#pragma unroll
        for (int i = 0; i < 8; ++i) {
            unsigned long long v = s[i];
            if (v) {
                lv[i]  = unfkey((unsigned int)(v >> 32));
                eid[i] = (int)(~(unsigned int)(v & 0xffffffffu));
            } else { lv[i] = 0.f; eid[i] = -1; }
        }
        float mmax = lv[0];            // cascade leaves slots sorted descending
        float a[8]; float den = 0.f;
#pragma unroll
        for (int i = 0; i < 8; ++i) {
            float ai = (eid[i] >= 0) ? __expf(lv[i] - mmax) : 0.f; // TEMP = 1
            a[i] = ai; den += ai;
        }
        float inv = 1.f / (den + 1e-6f);
        const int* eb = edges + (long long)b * 2 * E_;
#pragma unroll
        for (int i = 0; i < 8; ++i) {
            sw[g][i]   = a[i] * inv;
            ssrc[g][i] = (eid[i] >= 0) ? eb[eid[i]] : -1;   // src node of kept edge
        }
    }
    __syncthreads();
    if (active) {
        float acc = x[n * C_ + c];     // residual
#pragma unroll
        for (int i = 0; i < 8; ++i) {
            int si = ssrc[g][i];
            if (si >= 0) acc += sw[g][i] * y[((long long)b * K_ + si) * C_ + c];
        }
        out[n * C_ + c] = acc;
    }
}

extern "C" void kernel_launch(void* const* d_in, const int* in_sizes, int n_in,
                              void* d_out, int out_size, void* d_ws, size_t ws_size,
                              hipStream_t stream) {
    const float* node_feat = (const float*)d_in[0];   // [B,K,C] f32
    const int*   edges     = (const int*)d_in[1];     // [B,2,E] i32
    const float* logits    = (const float*)d_in[2];   // [B,E]   f32
    const int*   ntime     = (const int*)d_in[3];     // [B,K]   i32
    const float* W         = (const float*)d_in[4];   // [C,C]   f32
    float*       out       = (float*)d_out;           // [B,K,C] f32

    // workspace layout: y (B*K*C f32, 40.96MB) | slots (B*K*8 u64, 10.24MB)
    float* y = (float*)d_ws;
    unsigned long long* slots =
        (unsigned long long*)((char*)d_ws + (size_t)B_ * K_ * C_ * sizeof(float));

    long long nslots = (long long)B_ * K_ * 8;
    zero_slots_kernel<<<(int)((nslots + 255) / 256), 256, 0, stream>>>(slots, nslots);

    gemm_phi_kernel<<<B_ * 625, 256, 0, stream>>>(node_feat, W, y);

    long long ne = (long long)B_ * E_;
    edge_topk_kernel<<<(int)((ne + 255) / 256), 256, 0, stream>>>(edges, logits, ntime, slots);

    long long nn = (long long)B_ * K_;
    aggregate_kernel<<<(int)((nn + 3) / 4), 256, 0, stream>>>(node_feat, edges, slots, y, out);
}